// completeNet_82824149336898
// MI455X (gfx1250) — compile-verified
//
#include <hip/hip_runtime.h>
#include <math.h>

#define T_TRACK 256
#define M_DET   256
#define N_NODES 512
#define D_IN    512
#define D_EMB   256
#define D_APP   128

typedef __attribute__((ext_vector_type(2))) float v2f;
typedef __attribute__((ext_vector_type(8))) float v8f;

__device__ __forceinline__ float sigm(float z) { return 1.0f / (1.0f + expf(-z)); }

// D(16x16) += A(16xK) * B(Kx16), fp32 WMMA (V_WMMA_F32_16X16X4_F32).
// A element (i,k) at A[i*sAi + k*sAk]; B element (k,n) at B[k*sBk + n*sBn].
// One wave computes one tile; EXEC must be all ones (no divergence here).
__device__ __forceinline__ v8f wmma_f32_tile(const float* __restrict__ A, int sAi, int sAk,
                                             const float* __restrict__ B, int sBk, int sBn,
                                             int m0, int n0, int K, v8f acc) {
  const int lane = threadIdx.x & 31;
  const int l    = lane & 15;
  const int hh   = (lane >> 4) << 1;   // 0 or 2 (K sub-offset for upper half-wave)
  const float* Ap = A + (m0 + l) * sAi;
  const float* Bp = B + (n0 + l) * sBn;
  #pragma unroll 4
  for (int k = 0; k < K; k += 4) {
    v2f a, b;
    a.x = Ap[(k + hh)     * sAk];
    a.y = Ap[(k + hh + 1) * sAk];
    b.x = Bp[(k + hh)     * sBk];
    b.y = Bp[(k + hh + 1) * sBk];
    acc = __builtin_amdgcn_wmma_f32_16x16x4_f32(false, a, false, b, (short)0, acc, false, false);
  }
  return acc;
}

// ---------------- kernel 1: h = relu(x @ W_cnn + b_cnn)   [512,256] ----------------
__global__ void k_h(const float* __restrict__ x, const float* __restrict__ Wc,
                    const float* __restrict__ bc, float* __restrict__ h) {
  const int wave = (blockIdx.x * blockDim.x + threadIdx.x) >> 5;  // 512 waves
  const int m0 = (wave >> 4) << 4;   // 32 M-tiles
  const int n0 = (wave & 15) << 4;   // 16 N-tiles
  v8f acc = {};
  acc = wmma_f32_tile(x, D_IN, 1, Wc, D_EMB, 1, m0, n0, D_IN, acc);
  const int l  = threadIdx.x & 15;
  const int mh = ((threadIdx.x >> 4) & 1) << 3;
  const float bias = bc[n0 + l];
  #pragma unroll
  for (int r = 0; r < 8; ++r)
    h[(m0 + mh + r) * D_EMB + n0 + l] = fmaxf(acc[r] + bias, 0.0f);
}

// ---------------- kernel 2: A = h@Wapp1[:256], B = h@Wapp1[256:]  [512,128] each ----
__global__ void k_ab(const float* __restrict__ h, const float* __restrict__ Wapp1,
                     float* __restrict__ Aout, float* __restrict__ Bout) {
  const int wave  = (blockIdx.x * blockDim.x + threadIdx.x) >> 5;  // 512 waves
  const int which = wave >> 8;             // 0 -> Aout, 1 -> Bout
  const int t  = wave & 255;               // 32 x 8 tiles
  const int m0 = (t >> 3) << 4;
  const int n0 = (t & 7)  << 4;
  const float* Bw = Wapp1 + which * (D_EMB * D_APP);
  v8f acc = {};
  acc = wmma_f32_tile(h, D_EMB, 1, Bw, D_APP, 1, m0, n0, D_EMB, acc);
  float* dst = which ? Bout : Aout;
  const int l  = threadIdx.x & 15;
  const int mh = ((threadIdx.x >> 4) & 1) << 3;
  #pragma unroll
  for (int r = 0; r < 8; ++r)
    dst[(m0 + mh + r) * D_APP + n0 + l] = acc[r];
}

// ---------------- kernel 3: node geo partials gs = coords@Wg1[:4], gd = coords@Wg1[4:]
__global__ void k_geo(const float* __restrict__ coords, const float* __restrict__ Wg1,
                      float* __restrict__ gs, float* __restrict__ gd) {
  const int idx = blockIdx.x * blockDim.x + threadIdx.x;   // 512*32
  if (idx >= N_NODES * 32) return;
  const int node = idx >> 5, c = idx & 31;
  float s = 0.f, d = 0.f;
  #pragma unroll
  for (int k = 0; k < 4; ++k) {
    const float cv = coords[node * 4 + k];
    s += cv * Wg1[k * 32 + c];
    d += cv * Wg1[(4 + k) * 32 + c];
  }
  gs[idx] = s; gd[idx] = d;
}

// ---------------- kernel 4: per edge-pair (i,j): e1, e2, iou ------------------------
__global__ void k_edge(const float* __restrict__ Aout, const float* __restrict__ Bout,
                       const float* __restrict__ gs, const float* __restrict__ gd,
                       const float* __restrict__ cbox,
                       const float* __restrict__ bapp1, const float* __restrict__ Wapp2,
                       const float* __restrict__ bapp2,
                       const float* __restrict__ bg1, const float* __restrict__ Wg2,
                       const float* __restrict__ bg2,
                       const float* __restrict__ Wa1, const float* __restrict__ ba1,
                       const float* __restrict__ Wa2, const float* __restrict__ ba2,
                       float* __restrict__ e1, float* __restrict__ e2, float* __restrict__ iou1) {
  __shared__ float sAi[D_APP], sBi[D_APP], sb1[D_APP], sW2[D_APP];
  __shared__ float sgs[32], sgd[32], sbg1[32], sWg2[32];
  __shared__ float sbox[4], sWa1[16], sba1[8], sWa2[8];
  const int i = blockIdx.x;          // track index
  const int j = threadIdx.x;         // det index
  if (j < D_APP) {
    sAi[j] = Aout[i * D_APP + j];
    sBi[j] = Bout[i * D_APP + j];
    sb1[j] = bapp1[j];
    sW2[j] = Wapp2[j];
  }
  if (j < 32) { sgs[j] = gs[i * 32 + j]; sgd[j] = gd[i * 32 + j]; sbg1[j] = bg1[j]; sWg2[j] = Wg2[j]; }
  if (j < 16) sWa1[j] = Wa1[j];
  if (j < 8)  { sba1[j] = ba1[j]; sWa2[j] = Wa2[j]; }
  if (j < 4)  sbox[j] = cbox[i * 4 + j];
  __syncthreads();

  const int dn = T_TRACK + j;
  const float* Adet = Aout + dn * D_APP;
  const float* Bdet = Bout + dn * D_APP;
  const float b2 = bapp2[0];
  float z1 = b2, z2 = b2;
  #pragma unroll 8
  for (int c = 0; c < D_APP; ++c) {
    z1 += fmaxf(sAi[c] + Bdet[c] + sb1[c], 0.f) * sW2[c];   // edge track->det
    z2 += fmaxf(Adet[c] + sBi[c] + sb1[c], 0.f) * sW2[c];   // edge det->track
  }
  const float app1 = sigm(z1), app2 = sigm(z2);

  const float* gsd = gs + dn * 32;
  const float* gdd = gd + dn * 32;
  const float gb2 = bg2[0];
  float g1 = gb2, g2 = gb2;
  #pragma unroll
  for (int c = 0; c < 32; ++c) {
    g1 += fmaxf(sgs[c] + gdd[c] + sbg1[c], 0.f) * sWg2[c];
    g2 += fmaxf(gsd[c] + sgd[c] + sbg1[c], 0.f) * sWg2[c];
  }
  const float geo1 = sigm(g1), geo2 = sigm(g2);

  const float* bb = cbox + dn * 4;
  const float ix1 = fmaxf(sbox[0], bb[0]), iy1 = fmaxf(sbox[1], bb[1]);
  const float ix2 = fminf(sbox[2], bb[2]), iy2 = fminf(sbox[3], bb[3]);
  const float inter = fmaxf(ix2 - ix1, 0.f) * fmaxf(iy2 - iy1, 0.f);
  const float areaA = (sbox[2] - sbox[0]) * (sbox[3] - sbox[1]);
  const float areaB = (bb[2] - bb[0]) * (bb[3] - bb[1]);
  const float iouv = inter / (areaA + areaB - inter + 1e-8f);

  const float ab2 = ba2[0];
  float za = ab2, zb = ab2;
  #pragma unroll
  for (int k = 0; k < 8; ++k) {
    za += fmaxf(app1 * sWa1[k] + geo1 * sWa1[8 + k] + sba1[k], 0.f) * sWa2[k];
    zb += fmaxf(app2 * sWa1[k] + geo2 * sWa1[8 + k] + sba1[k], 0.f) * sWa2[k];
  }
  const int o = i * M_DET + j;
  e1[o] = sigm(za);
  e2[o] = sigm(zb);
  iou1[o] = iouv;
}

// ---------------- kernel 5: agg via two 256^3 GEMMs --------------------------------
// rows 0..255 (tracks):  agg = E2 @ H_det ; rows 256..511 (dets): agg = E1^T @ H_trk
__global__ void k_agg(const float* __restrict__ e1, const float* __restrict__ e2,
                      const float* __restrict__ h, float* __restrict__ agg) {
  const int wave = (blockIdx.x * blockDim.x + threadIdx.x) >> 5;  // 512 waves
  const int m0 = (wave >> 4) << 4;
  const int n0 = (wave & 15) << 4;
  v8f acc = {};
  if (m0 < T_TRACK)
    acc = wmma_f32_tile(e2, M_DET, 1, h + T_TRACK * D_EMB, D_EMB, 1, m0, n0, M_DET, acc);
  else  // A(j,k) = e1[k*256 + j]  (transposed read, no materialized transpose)
    acc = wmma_f32_tile(e1, 1, M_DET, h, D_EMB, 1, m0 - T_TRACK, n0, T_TRACK, acc);
  const int l  = threadIdx.x & 15;
  const int mh = ((threadIdx.x >> 4) & 1) << 3;
  #pragma unroll
  for (int r = 0; r < 8; ++r)
    agg[(m0 + mh + r) * D_EMB + n0 + l] = acc[r];
}

// ---------------- kernel 6: out = relu(h@Wo1 + agg@Wo2 + bo) -----------------------
__global__ void k_out(const float* __restrict__ h, const float* __restrict__ agg,
                      const float* __restrict__ Wo1, const float* __restrict__ Wo2,
                      const float* __restrict__ bo, float* __restrict__ outm) {
  const int wave = (blockIdx.x * blockDim.x + threadIdx.x) >> 5;  // 512 waves
  const int m0 = (wave >> 4) << 4;
  const int n0 = (wave & 15) << 4;
  v8f acc = {};
  acc = wmma_f32_tile(h,   D_EMB, 1, Wo1, D_EMB, 1, m0, n0, D_EMB, acc);
  acc = wmma_f32_tile(agg, D_EMB, 1, Wo2, D_EMB, 1, m0, n0, D_EMB, acc);
  const int l  = threadIdx.x & 15;
  const int mh = ((threadIdx.x >> 4) & 1) << 3;
  const float bias = bo[n0 + l];
  #pragma unroll
  for (int r = 0; r < 8; ++r)
    outm[(m0 + mh + r) * D_EMB + n0 + l] = fmaxf(acc[r] + bias, 0.0f);
}

// ---------------- kernel 7: row norms of out ---------------------------------------
__global__ void k_norm(const float* __restrict__ outm, float* __restrict__ nrm) {
  const int n = blockIdx.x * blockDim.x + threadIdx.x;
  if (n >= N_NODES) return;
  const float* row = outm + n * D_EMB;
  float s = 0.f;
  #pragma unroll 8
  for (int c = 0; c < D_EMB; ++c) s += row[c] * row[c];
  nrm[n] = fmaxf(sqrtf(s), 1e-6f);
}

// ---------------- kernel 8: dots GEMM + cosine + final MLP + exp(5*score) ----------
__global__ void k_cos(const float* __restrict__ outm, const float* __restrict__ nrm,
                      const float* __restrict__ iou1,
                      const float* __restrict__ Wf1, const float* __restrict__ bf1,
                      const float* __restrict__ Wf2, const float* __restrict__ bf2,
                      float* __restrict__ Mval) {
  const int wave = (blockIdx.x * blockDim.x + threadIdx.x) >> 5;  // 256 waves
  const int m0 = (wave >> 4) << 4;
  const int n0 = (wave & 15) << 4;
  // B(k,j) = out_det[j,k] -> transposed read of det rows
  v8f acc = {};
  acc = wmma_f32_tile(outm, D_EMB, 1, outm + T_TRACK * D_EMB, 1, D_EMB, m0, n0, D_EMB, acc);
  const int l  = threadIdx.x & 15;
  const int mh = ((threadIdx.x >> 4) & 1) << 3;
  const int j  = n0 + l;
  const float nb = nrm[T_TRACK + j];
  const float fb2 = bf2[0];
  #pragma unroll
  for (int r = 0; r < 8; ++r) {
    const int i = m0 + mh + r;
    const float cosv = acc[r] / (nrm[i] * nb);
    const float io = iou1[i * M_DET + j];
    float z = fb2;
    #pragma unroll
    for (int k = 0; k < 8; ++k)
      z += fmaxf(cosv * Wf1[k] + io * Wf1[8 + k] + bf1[k], 0.f) * Wf2[k];
    Mval[i * M_DET + j] = expf(5.0f * sigm(z));   // LAM = 5
  }
}

// ---------------- kernel 9: Sinkhorn, whole 257x257 matrix in LDS ------------------
__global__ void k_sink(const float* __restrict__ Mval, float* __restrict__ outp) {
  __shared__ float smat[257 * 257];   // 264196 B of the 320KB WGP LDS
  const int tid = threadIdx.x, lane = tid & 31, wid = tid >> 5;
  const float slackv = expf(5.0f * 0.2f);   // exp(LAM*SLACK)
  for (int idx = tid; idx < 257 * 257; idx += 1024) {
    const int r = idx / 257, c = idx % 257;
    smat[idx] = (r < 256 && c < 256) ? Mval[r * 256 + c] : slackv;
  }
  __syncthreads();
  for (int it = 0; it < 8; ++it) {
    // row normalize: one wave per row, lanes stride columns (bank-friendly)
    for (int r = wid; r < 257; r += 32) {
      float s = 0.f;
      for (int c = lane; c < 257; c += 32) s += smat[r * 257 + c];
      #pragma unroll
      for (int off = 16; off > 0; off >>= 1) s += __shfl_xor(s, off, 32);
      for (int c = lane; c < 257; c += 32) smat[r * 257 + c] /= s;
    }
    __syncthreads();
    // col normalize: stride 257 == 1 (mod 64 banks) -> conflict-free
    for (int c = wid; c < 257; c += 32) {
      float s = 0.f;
      for (int r = lane; r < 257; r += 32) s += smat[r * 257 + c];
      #pragma unroll
      for (int off = 16; off > 0; off >>= 1) s += __shfl_xor(s, off, 32);
      for (int r = lane; r < 257; r += 32) smat[r * 257 + c] /= s;
    }
    __syncthreads();
  }
  for (int idx = tid; idx < 256 * 256; idx += 1024) {
    const int r = idx >> 8, c = idx & 255;
    outp[idx] = smat[r * 257 + c];
  }
}

extern "C" void kernel_launch(void* const* d_in, const int* in_sizes, int n_in,
                              void* d_out, int out_size, void* d_ws, size_t ws_size,
                              hipStream_t stream) {
  (void)in_sizes; (void)n_in; (void)out_size; (void)ws_size;
  const float* x      = (const float*)d_in[0];
  const float* coords = (const float*)d_in[1];
  const float* cbox   = (const float*)d_in[2];
  /* d_in[3] edge_index: fixed full-bipartite structure, handled analytically */
  const float* Wc     = (const float*)d_in[4];
  const float* bc     = (const float*)d_in[5];
  const float* Wapp1  = (const float*)d_in[6];
  const float* bapp1  = (const float*)d_in[7];
  const float* Wapp2  = (const float*)d_in[8];
  const float* bapp2  = (const float*)d_in[9];
  const float* Wg1    = (const float*)d_in[10];
  const float* bg1    = (const float*)d_in[11];
  const float* Wg2    = (const float*)d_in[12];
  const float* bg2    = (const float*)d_in[13];
  const float* Wa1    = (const float*)d_in[14];
  const float* ba1    = (const float*)d_in[15];
  const float* Wa2    = (const float*)d_in[16];
  const float* ba2    = (const float*)d_in[17];
  const float* Wo1    = (const float*)d_in[18];
  const float* Wo2    = (const float*)d_in[19];
  const float* bo     = (const float*)d_in[20];
  const float* Wf1    = (const float*)d_in[21];
  const float* bf1    = (const float*)d_in[22];
  const float* Wf2    = (const float*)d_in[23];
  const float* bf2    = (const float*)d_in[24];

  float* ws   = (float*)d_ws;
  float* h    = ws;               // 131072
  float* Aout = h    + 131072;    // 65536
  float* Bout = Aout + 65536;     // 65536
  float* gs   = Bout + 65536;     // 16384
  float* gd   = gs   + 16384;     // 16384
  float* e1   = gd   + 16384;     // 65536
  float* e2   = e1   + 65536;     // 65536
  float* iou1 = e2   + 65536;     // 65536
  float* agg  = iou1 + 65536;     // 131072
  float* outm = agg  + 131072;    // 131072
  float* nrm  = outm + 131072;    // 512
  float* Mval = nrm  + 512;       // 65536

  k_h   <<<128, 128, 0, stream>>>(x, Wc, bc, h);
  k_ab  <<<128, 128, 0, stream>>>(h, Wapp1, Aout, Bout);
  k_geo <<<64, 256, 0, stream>>>(coords, Wg1, gs, gd);
  k_edge<<<256, 256, 0, stream>>>(Aout, Bout, gs, gd, cbox, bapp1, Wapp2, bapp2,
                                  bg1, Wg2, bg2, Wa1, ba1, Wa2, ba2, e1, e2, iou1);
  k_agg <<<128, 128, 0, stream>>>(e1, e2, h, agg);
  k_out <<<128, 128, 0, stream>>>(h, agg, Wo1, Wo2, bo, outm);
  k_norm<<<2, 256, 0, stream>>>(outm, nrm);
  k_cos <<<64, 128, 0, stream>>>(outm, nrm, iou1, Wf1, bf1, Wf2, bf2, Mval);
  k_sink<<<1, 1024, 0, stream>>>(Mval, (float*)d_out);
}